// GraphAttentionNetwork_49014166782293
// MI455X (gfx1250) — compile-verified
//
#include <hip/hip_runtime.h>

typedef __attribute__((ext_vector_type(2))) float v2f;
typedef __attribute__((ext_vector_type(8))) float v8f;

#define DHID 128
#define NHEAD 4
#define CHN 32
#define KCHUNK 32
#define WSTRIDE 144   // 128 + 16 pad: 2*144 % 64 == 32 -> lane-half bank groups disjoint

// float atomic max via signed-int max / unsigned-int min (valid with -inf init)
__device__ __forceinline__ void atomicMaxF(float* addr, float v) {
  if (v >= 0.0f) atomicMax((int*)addr, __float_as_int(v));
  else           atomicMin((unsigned int*)addr, __float_as_uint(v));
}

__global__ void k_zero(float* p, int n) {
  int i = blockIdx.x * blockDim.x + threadIdx.x;
  if (i < n) p[i] = 0.0f;
}

// S[d] += edge_attr[e]; cnt[d] += 1   (for self-loop mean attr)
__global__ void k_edge_stats(const int* __restrict__ dst, const float* __restrict__ eattr,
                             float* __restrict__ S, float* __restrict__ cnt, int E) {
  int e = blockIdx.x * blockDim.x + threadIdx.x;
  if (e >= E) return;
  int d = dst[e];
  atomicAdd(&S[d], eattr[e]);
  atomicAdd(&cnt[d], 1.0f);
}

// p[l,h] = sum_j ew[j] * v[l,j,h],  q[l,h] = sum_j eb[j] * v[l,j,h]
// v[l,j,h] = sum_c lin_edge_w[l][j, h*32+c] * att_edge[l,h,c]
__global__ void k_pq(const float* __restrict__ lin_edge_w, const float* __restrict__ att_edge,
                     const float* __restrict__ ew, const float* __restrict__ eb,
                     float* __restrict__ pq) {
  int t = threadIdx.x;
  if (t >= 12) return;
  int l = t >> 2, h = t & 3;
  const float* W = lin_edge_w + l * DHID * DHID;
  const float* a = att_edge + l * DHID + h * CHN;
  float p = 0.0f, q = 0.0f;
  for (int j = 0; j < DHID; ++j) {
    float v = 0.0f;
    for (int c = 0; c < CHN; ++c) v += W[j * DHID + h * CHN + c] * a[c];
    p += ew[j] * v;
    q += eb[j] * v;
  }
  pq[t] = p;
  pq[12 + t] = q;
}

// node encoder: h = x[N,17] @ W[17,128] + b   (tiny: 0.2 GFLOP, scalar is fine)
__global__ void k_encode(const float* __restrict__ x, const float* __restrict__ w,
                         const float* __restrict__ b, float* __restrict__ h, int N) {
  int i = blockIdx.x * blockDim.x + threadIdx.x;
  if (i >= N * DHID) return;
  int n = i >> 7, j = i & 127;
  float s = b[j];
  #pragma unroll
  for (int k = 0; k < 17; ++k) s += x[n * 17 + k] * w[k * DHID + j];
  h[i] = s;
}

// fp32 WMMA GEMM: out[N,128] = A[N,128] @ W[128,128] (+bias, optional relu).
// 256-thread block = 8 waves; each wave owns a 16-row x 128-col tile.
// W is staged through LDS in 32x128 K-chunks (bank-conflict-free padded stride),
// shared by all 8 waves -> 8x less global B traffic than per-wave streaming.
// No early returns: barriers must be block-uniform and WMMA needs EXEC all-ones;
// out-of-range rows clamp their A reads and predicate only the stores.
__global__ __launch_bounds__(256) void k_gemm128_wmma(
    const float* __restrict__ A, const float* __restrict__ W,
    float* __restrict__ out, int N,
    const float* __restrict__ bias, int relu) {
  __shared__ float Wlds[KCHUNK * WSTRIDE];   // 18 KB of the WGP's 320 KB LDS

  int lane = threadIdx.x & 31;
  int waveInBlk = threadIdx.x >> 5;
  int row0 = (blockIdx.x * 8 + waveInBlk) * 16;
  int half = lane >> 4;                      // 0: lanes 0-15 (K pair 0,1), 1: lanes 16-31 (K pair 2,3)
  int lr = lane & 15;
  int rowA = row0 + lr;
  if (rowA >= N) rowA = N - 1;               // clamp; D-row m only depends on A-row m
  const float* arow = A + (size_t)rowA * DHID;

  v8f zero = {};
  v8f acc[8];
  #pragma unroll
  for (int t = 0; t < 8; ++t) acc[t] = zero;

  for (int k0 = 0; k0 < DHID; k0 += KCHUNK) {
    // stage W[k0..k0+31][0..127] into LDS with padded rows
    for (int t = threadIdx.x; t < KCHUNK * DHID; t += 256) {
      int r = t >> 7, c = t & 127;
      Wlds[r * WSTRIDE + c] = W[(size_t)(k0 + r) * DHID + c];
    }
    __syncthreads();

    #pragma unroll
    for (int kk = 0; kk < KCHUNK; kk += 4) {
      int ka = kk + 2 * half;
      v2f a;                                 // A[16x4] fragment
      a.x = arow[k0 + ka];
      a.y = arow[k0 + ka + 1];
      #pragma unroll
      for (int t = 0; t < 8; ++t) {
        int col = t * 16 + lr;
        v2f b;                               // B[4x16] fragment from LDS
        b.x = Wlds[ka * WSTRIDE + col];
        b.y = Wlds[(ka + 1) * WSTRIDE + col];
        acc[t] = __builtin_amdgcn_wmma_f32_16x16x4_f32(
            false, a, false, b, (short)0, acc[t], false, false);
      }
    }
    __syncthreads();
  }

  #pragma unroll
  for (int t = 0; t < 8; ++t) {
    int col = t * 16 + lr;
    float bb = bias ? bias[col] : 0.0f;
    #pragma unroll
    for (int v = 0; v < 8; ++v) {            // D layout: VGPR v -> row v (lanes 0-15) / v+8 (lanes 16-31)
      int row = row0 + v + 8 * half;
      float val = acc[t][v] + bb;
      if (relu) val = fmaxf(val, 0.0f);
      if (row < N) out[(size_t)row * DHID + col] = val;
    }
  }
}

// zero accumulator; init per-(node,head) max=-inf, den=0
__global__ void k_prep(float* __restrict__ acc, float* __restrict__ mx,
                       float* __restrict__ den, int N) {
  int i = blockIdx.x * blockDim.x + threadIdx.x;
  if (i < N * DHID) acc[i] = 0.0f;
  if (i < N * NHEAD) { mx[i] = -__builtin_inff(); den[i] = 0.0f; }
}

// alpha_src[n,h] = <hp[n,h,:], att_src[h,:]>, same for dst
__global__ void k_alpha_node(const float* __restrict__ hp, const float* __restrict__ ws,
                             const float* __restrict__ wd, float* __restrict__ asrc,
                             float* __restrict__ adst, int N) {
  int i = blockIdx.x * blockDim.x + threadIdx.x;
  if (i >= N * NHEAD) return;
  int n = i >> 2, h = i & 3;
  const float* row = hp + (size_t)n * DHID + h * CHN;
  const float* w1 = ws + h * CHN;
  const float* w2 = wd + h * CHN;
  float s1 = 0.0f, s2 = 0.0f;
  #pragma unroll
  for (int c = 0; c < CHN; ++c) { float v = row[c]; s1 += v * w1[c]; s2 += v * w2[c]; }
  asrc[i] = s1;
  adst[i] = s2;
}

// logits = leaky_relu(asrc[src]+adst[dst]+alpha_e), segment-max into mx[dst]
__global__ void k_edge_logits(const int* __restrict__ src, const int* __restrict__ dst,
                              const float* __restrict__ eattr,
                              const float* __restrict__ asrc, const float* __restrict__ adst,
                              const float* __restrict__ S, const float* __restrict__ cnt,
                              const float* __restrict__ pq, int l,
                              float* __restrict__ mx, float* __restrict__ logits,
                              int E, int N) {
  int e = blockIdx.x * blockDim.x + threadIdx.x;
  if (e >= E + N) return;
  const float* p = pq + l * 4;
  const float* q = pq + 12 + l * 4;
  int s, d;
  float ae[NHEAD];
  if (e < E) {
    s = src[e]; d = dst[e];
    float a = eattr[e];
    #pragma unroll
    for (int h = 0; h < NHEAD; ++h) ae[h] = a * p[h] + q[h];
  } else {
    s = d = e - E;                       // self-loop, attr = mean of incoming edge attrs
    float c = cnt[s], m = fmaxf(c, 1.0f), Sv = S[s];
    #pragma unroll
    for (int h = 0; h < NHEAD; ++h) ae[h] = (Sv * p[h] + c * q[h]) / m;
  }
  #pragma unroll
  for (int h = 0; h < NHEAD; ++h) {
    float lg = asrc[s * 4 + h] + adst[d * 4 + h] + ae[h];
    lg = lg > 0.0f ? lg : 0.2f * lg;     // leaky_relu(0.2)
    logits[(size_t)e * 4 + h] = lg;
    atomicMaxF(&mx[d * 4 + h], lg);
  }
}

// num = exp(logit - mx[dst]); den[dst] += num   (num overwrites logits in place)
__global__ void k_softmax_num(const int* __restrict__ dst, float* __restrict__ logits,
                              const float* __restrict__ mx, float* __restrict__ den,
                              int E, int N) {
  int e = blockIdx.x * blockDim.x + threadIdx.x;
  if (e >= E + N) return;
  int d = (e < E) ? dst[e] : e - E;
  #pragma unroll
  for (int h = 0; h < NHEAD; ++h) {
    float nm = __expf(logits[(size_t)e * 4 + h] - mx[d * 4 + h]);
    logits[(size_t)e * 4 + h] = nm;
    atomicAdd(&den[d * 4 + h], nm);
  }
}

// acc[dst] += hp[src] * (num/(den[dst]+eps));  one thread per (edge, 4-float chunk)
__global__ void k_scatter(const int* __restrict__ src, const int* __restrict__ dst,
                          const float* __restrict__ num, const float* __restrict__ den,
                          const float* __restrict__ hp, float* __restrict__ acc,
                          int E, int N) {
  int tid = blockIdx.x * blockDim.x + threadIdx.x;
  int total = (E + N) * 32;
  if (tid >= total) return;
  int e = tid >> 5;
  int j4 = tid & 31;
  int j = j4 << 2;
  int h = j >> 5;
  int s, d;
  if (e < E) { s = src[e]; d = dst[e]; } else { s = d = e - E; }
  float alpha = num[(size_t)e * 4 + h] / (den[d * 4 + h] + 1e-16f);
  const float4* hp4 = (const float4*)(hp + (size_t)s * DHID);
  float4 v = hp4[j4];
  float* o = acc + (size_t)d * DHID + j;
  atomicAdd(o + 0, v.x * alpha);
  atomicAdd(o + 1, v.y * alpha);
  atomicAdd(o + 2, v.z * alpha);
  atomicAdd(o + 3, v.w * alpha);
}

// h = elu(acc + gat_bias)
__global__ void k_bias_elu(const float* __restrict__ acc, const float* __restrict__ bias,
                           float* __restrict__ h, int N) {
  int i = blockIdx.x * blockDim.x + threadIdx.x;
  if (i >= N * DHID) return;
  float v = acc[i] + bias[i & 127];
  h[i] = v > 0.0f ? v : (__expf(v) - 1.0f);
}

// out[N,2] = hd @ dec_w2 + dec_b2
__global__ void k_dec2(const float* __restrict__ hd, const float* __restrict__ w2,
                       const float* __restrict__ b2, float* __restrict__ out, int N) {
  int n = blockIdx.x * blockDim.x + threadIdx.x;
  if (n >= N) return;
  float s0 = b2[0], s1 = b2[1];
  const float* row = hd + (size_t)n * DHID;
  #pragma unroll 8
  for (int j = 0; j < DHID; ++j) {
    float v = row[j];
    s0 += v * w2[j * 2 + 0];
    s1 += v * w2[j * 2 + 1];
  }
  out[n * 2 + 0] = s0;
  out[n * 2 + 1] = s1;
}

extern "C" void kernel_launch(void* const* d_in, const int* in_sizes, int n_in,
                              void* d_out, int out_size, void* d_ws, size_t ws_size,
                              hipStream_t stream) {
  (void)n_in; (void)out_size; (void)ws_size;
  const float* x          = (const float*)d_in[0];
  const int*   ei         = (const int*)d_in[1];
  const float* eattr      = (const float*)d_in[2];
  const float* enc_w      = (const float*)d_in[3];
  const float* enc_b      = (const float*)d_in[4];
  const float* ew         = (const float*)d_in[5];
  const float* eb         = (const float*)d_in[6];
  const float* lin_w      = (const float*)d_in[7];
  const float* lin_edge_w = (const float*)d_in[8];
  const float* att_src    = (const float*)d_in[9];
  const float* att_dst    = (const float*)d_in[10];
  const float* att_edge   = (const float*)d_in[11];
  const float* gat_bias   = (const float*)d_in[12];
  const float* dec_w1     = (const float*)d_in[13];
  const float* dec_b1     = (const float*)d_in[14];
  const float* dec_w2     = (const float*)d_in[15];
  const float* dec_b2     = (const float*)d_in[16];
  float* out = (float*)d_out;

  const int N  = in_sizes[0] / 17;
  const int E  = in_sizes[1] / 2;
  const int E2 = E + N;
  const int* srcIdx = ei;
  const int* dstIdx = ei + E;

  // workspace carve-out (256B aligned)
  char* base = (char*)d_ws;
  size_t off = 0;
  auto alloc = [&](size_t bytes) -> void* {
    off = (off + 255) & ~(size_t)255;
    void* p = base + off;
    off += bytes;
    return p;
  };
  float* h      = (float*)alloc((size_t)N * DHID * 4);
  float* hp     = (float*)alloc((size_t)N * DHID * 4);
  float* acc    = (float*)alloc((size_t)N * DHID * 4);
  float* S      = (float*)alloc((size_t)2 * N * 4);   // S then cnt, contiguous
  float* cnt    = S + N;
  float* asrc   = (float*)alloc((size_t)N * NHEAD * 4);
  float* adst   = (float*)alloc((size_t)N * NHEAD * 4);
  float* mx     = (float*)alloc((size_t)N * NHEAD * 4);
  float* den    = (float*)alloc((size_t)N * NHEAD * 4);
  float* logits = (float*)alloc((size_t)E2 * NHEAD * 4);
  float* pq     = (float*)alloc(24 * 4);

  const int B = 256;
  auto g = [](int n, int b) { return (n + b - 1) / b; };
  int gemmBlocks = (N + 127) / 128;   // 8 waves/block, 16 rows/wave

  k_zero<<<g(2 * N, B), B, 0, stream>>>(S, 2 * N);
  k_edge_stats<<<g(E, B), B, 0, stream>>>(dstIdx, eattr, S, cnt, E);
  k_pq<<<1, 32, 0, stream>>>(lin_edge_w, att_edge, ew, eb, pq);
  k_encode<<<g(N * DHID, B), B, 0, stream>>>(x, enc_w, enc_b, h, N);

  for (int l = 0; l < 3; ++l) {
    k_gemm128_wmma<<<gemmBlocks, B, 0, stream>>>(h, lin_w + l * DHID * DHID, hp, N,
                                                 nullptr, 0);
    k_prep<<<g(N * DHID, B), B, 0, stream>>>(acc, mx, den, N);
    k_alpha_node<<<g(N * NHEAD, B), B, 0, stream>>>(hp, att_src + l * DHID, att_dst + l * DHID,
                                                    asrc, adst, N);
    k_edge_logits<<<g(E2, B), B, 0, stream>>>(srcIdx, dstIdx, eattr, asrc, adst, S, cnt,
                                              pq, l, mx, logits, E, N);
    k_softmax_num<<<g(E2, B), B, 0, stream>>>(dstIdx, logits, mx, den, E, N);
    k_scatter<<<g(E2 * 32, B), B, 0, stream>>>(srcIdx, dstIdx, logits, den, hp, acc, E, N);
    k_bias_elu<<<g(N * DHID, B), B, 0, stream>>>(acc, gat_bias + l * DHID, h, N);
  }

  k_gemm128_wmma<<<gemmBlocks, B, 0, stream>>>(h, dec_w1, hp, N, dec_b1, 1);
  k_dec2<<<g(N, B), B, 0, stream>>>(hp, dec_w2, dec_b2, out, N);
}